// GcnIdSimple_7919919694202
// MI455X (gfx1250) — compile-verified
//
#include <hip/hip_runtime.h>

#define N_NODES 200000
#define N_EDGES 6400000
#define F_IN 128
#define F_HID 64

typedef float v2f __attribute__((ext_vector_type(2)));
typedef float v8f __attribute__((ext_vector_type(8)));

// ---------------------------------------------------------------- utilities

__global__ void gcn_zero_kernel(float* __restrict__ p, long n) {
  long i = (long)blockIdx.x * blockDim.x + threadIdx.x;
  long stride = (long)gridDim.x * blockDim.x;
  for (; i < n; i += stride) p[i] = 0.0f;
}

__device__ __forceinline__ void atomic_add_f32(float* p, float v) {
  __hip_atomic_fetch_add(p, v, __ATOMIC_RELAXED, __HIP_MEMORY_SCOPE_AGENT);
}

// deg[col[e]] += w[e]
__global__ void gcn_deg_kernel(const long long* __restrict__ col,
                               const float* __restrict__ w,
                               float* __restrict__ deg, int E) {
  int e = blockIdx.x * blockDim.x + threadIdx.x;
  if (e < E) atomic_add_f32(&deg[(int)col[e]], w[e]);
}

// deg -> deg_inv_sqrt (in place)
__global__ void gcn_rsqrt_kernel(float* __restrict__ deg, int n) {
  int i = blockIdx.x * blockDim.x + threadIdx.x;
  if (i < n) {
    float d = deg[i];
    deg[i] = (d > 0.0f) ? __frsqrt_rn(d) : 0.0f;
  }
}

// norm[e] = dis[row[e]] * w[e] * dis[col[e]]
__global__ void gcn_norm_kernel(const long long* __restrict__ row,
                                const long long* __restrict__ col,
                                const float* __restrict__ w,
                                const float* __restrict__ dis,
                                float* __restrict__ norm, int E) {
  int e = blockIdx.x * blockDim.x + threadIdx.x;
  if (e < E) norm[e] = dis[(int)row[e]] * w[e] * dis[(int)col[e]];
}

// ---------------------------------------------------------------- WMMA GEMM
// out[M, NOUT] = A[M, K] @ W[NOUT, K]^T   (PyG Linear weight layout: [out,in])
//
// One wave32 owns one 16-row M-stripe and accumulates ALL NOUT/16 column
// tiles in registers, so each A slice is loaded exactly once and reused
// across NT WMMAs (NT=4 for layer 1, NT=8 for layer 2). A is then read once
// total (102 MB / 51 MB) instead of NT times. Exact f32 math via
// V_WMMA_F32_16X16X4_F32; GEMM compute is trivial vs the 10 GB edge traffic.
//
// f32 16x4 A layout (ISA 7.12.2): lane l holds row m=l&15; the two A VGPRs
// hold K = kb, kb+1 with kb = k + (l>>4)*2. B (4x16) mirrors with the lane
// indexing the N column. C/D: lane l = column, VGPR j = row j + (l>>4)*8.
template <int K, int NOUT>
__global__ void gcn_gemm_wmma_kernel(const float* __restrict__ A,
                                     const float* __restrict__ W,
                                     float* __restrict__ out, int M) {
  constexpr int NT = NOUT / 16;
  const int lane = threadIdx.x & 31;
  const int wave = threadIdx.x >> 5;
  const int mt = blockIdx.x * (blockDim.x >> 5) + wave;  // wave-uniform
  if (mt * 16 >= M) return;  // uniform across the wave -> EXEC stays all-1s

  const int half = lane >> 4;          // 0: lanes 0-15, 1: lanes 16-31
  const int mn = lane & 15;
  const float* __restrict__ arow = A + (long)(mt * 16 + mn) * K;

  v8f c[NT];
#pragma unroll
  for (int t = 0; t < NT; ++t) c[t] = (v8f){};

#pragma unroll
  for (int k = 0; k < K; k += 4) {
    const int kb = k + half * 2;                  // 8B-aligned (kb even)
    const v2f a = *(const v2f*)(arow + kb);       // A slice: load once...
#pragma unroll
    for (int t = 0; t < NT; ++t) {                // ...reuse across N tiles
      const v2f b = *(const v2f*)(W + (long)(t * 16 + mn) * K + kb);
      c[t] = __builtin_amdgcn_wmma_f32_16x16x4_f32(
          /*neg_a=*/false, a, /*neg_b=*/false, b,
          /*c_mod=*/(short)0, c[t], /*reuse_a=*/false, /*reuse_b=*/false);
    }
  }

  const int rbase = mt * 16 + half * 8;
#pragma unroll
  for (int t = 0; t < NT; ++t) {
#pragma unroll
    for (int j = 0; j < 8; ++j)
      out[(long)(rbase + j) * NOUT + t * 16 + mn] = c[t][j];
  }
}

// ---------------------------------------------------------------- scatter
// One wave per edge: coalesced row gather (float2 per lane = 256B row),
// scaled, scatter-added with native f32 atomics (resolve in L2; both the
// source and destination node-feature matrices fit in the 192MB L2).
__global__ void gcn_scatter64_kernel(const long long* __restrict__ row,
                                     const long long* __restrict__ col,
                                     const float* __restrict__ norm,
                                     const float* __restrict__ src,
                                     float* __restrict__ dst, int E) {
  int e = blockIdx.x * (blockDim.x >> 5) + (threadIdx.x >> 5);
  if (e >= E) return;
  const int lane = threadIdx.x & 31;
  const int r = (int)row[e];
  const int c = (int)col[e];
  const float w = norm[e];
  const float2 s = *(const float2*)(src + (long)r * 64 + lane * 2);
  float* d = dst + (long)c * 64 + lane * 2;
  atomic_add_f32(d + 0, w * s.x);
  atomic_add_f32(d + 1, w * s.y);
}

// 128-feature variant: float4 per lane = one 512B row per edge.
__global__ void gcn_scatter128_kernel(const long long* __restrict__ row,
                                      const long long* __restrict__ col,
                                      const float* __restrict__ norm,
                                      const float* __restrict__ src,
                                      float* __restrict__ dst, int E) {
  int e = blockIdx.x * (blockDim.x >> 5) + (threadIdx.x >> 5);
  if (e >= E) return;
  const int lane = threadIdx.x & 31;
  const int r = (int)row[e];
  const int c = (int)col[e];
  const float w = norm[e];
  const float4 s = *(const float4*)(src + (long)r * 128 + lane * 4);
  float* d = dst + (long)c * 128 + lane * 4;
  atomic_add_f32(d + 0, w * s.x);
  atomic_add_f32(d + 1, w * s.y);
  atomic_add_f32(d + 2, w * s.z);
  atomic_add_f32(d + 3, w * s.w);
}

// h[i] = relu(h[i] + b[i % F]), F a power of two
template <int F>
__global__ void gcn_bias_relu_kernel(float* __restrict__ h,
                                     const float* __restrict__ b, long total) {
  long i = (long)blockIdx.x * blockDim.x + threadIdx.x;
  if (i < total) {
    float v = h[i] + b[i & (F - 1)];
    h[i] = v > 0.0f ? v : 0.0f;
  }
}

// ---------------------------------------------------------------- launch

extern "C" void kernel_launch(void* const* d_in, const int* in_sizes, int n_in,
                              void* d_out, int out_size, void* d_ws, size_t ws_size,
                              hipStream_t stream) {
  const float* x = (const float*)d_in[0];                // [N, 128]
  const long long* eidx = (const long long*)d_in[1];     // [2, E] int64
  const float* ew = (const float*)d_in[2];               // [E]
  const float* W1 = (const float*)d_in[3];               // [64, 128]
  const float* b1 = (const float*)d_in[4];               // [64]
  const float* W2 = (const float*)d_in[5];               // [128, 64]
  const float* b2 = (const float*)d_in[6];               // [128]
  float* out = (float*)d_out;                            // [N, 128]

  const long long* row = eidx;            // sources
  const long long* col = eidx + N_EDGES;  // targets

  // workspace carve-out (~231 MB)
  float* ws = (float*)d_ws;
  float* deg = ws;                                 // N        (then deg^-1/2)
  float* norm = deg + N_NODES;                     // E
  float* lin1 = norm + N_EDGES;                    // N*64
  float* agg1 = lin1 + (long)N_NODES * F_HID;      // N*64  (h1 after relu)
  float* lin2 = agg1 + (long)N_NODES * F_HID;      // N*128

  const int BT = 256;

  // zero accumulators
  gcn_zero_kernel<<<1024, BT, 0, stream>>>(deg, (long)N_NODES);
  gcn_zero_kernel<<<4096, BT, 0, stream>>>(agg1, (long)N_NODES * F_HID);
  gcn_zero_kernel<<<4096, BT, 0, stream>>>(out, (long)N_NODES * F_IN);

  // degree -> inverse sqrt -> per-edge norm
  gcn_deg_kernel<<<(N_EDGES + BT - 1) / BT, BT, 0, stream>>>(col, ew, deg, N_EDGES);
  gcn_rsqrt_kernel<<<(N_NODES + BT - 1) / BT, BT, 0, stream>>>(deg, N_NODES);
  gcn_norm_kernel<<<(N_EDGES + BT - 1) / BT, BT, 0, stream>>>(row, col, ew, deg,
                                                              norm, N_EDGES);

  const int mtiles = (N_NODES + 15) / 16;           // 12500 (exact)
  const int mblocks = (mtiles + 7) / 8;             // 8 waves per 256-thr block

  // layer 1: lin1 = x @ W1^T  (M=200000, K=128, NOUT=64; 4 n-tiles per wave)
  gcn_gemm_wmma_kernel<F_IN, F_HID>
      <<<mblocks, BT, 0, stream>>>(x, W1, lin1, N_NODES);
  gcn_scatter64_kernel<<<(N_EDGES + 7) / 8, BT, 0, stream>>>(row, col, norm,
                                                             lin1, agg1, N_EDGES);
  gcn_bias_relu_kernel<F_HID>
      <<<(int)(((long)N_NODES * F_HID + BT - 1) / BT), BT, 0, stream>>>(
          agg1, b1, (long)N_NODES * F_HID);

  // layer 2: lin2 = h1 @ W2^T  (M=200000, K=64, NOUT=128; 8 n-tiles per wave)
  gcn_gemm_wmma_kernel<F_HID, F_IN>
      <<<mblocks, BT, 0, stream>>>(agg1, W2, lin2, N_NODES);
  gcn_scatter128_kernel<<<(N_EDGES + 7) / 8, BT, 0, stream>>>(row, col, norm,
                                                              lin2, out, N_EDGES);
  gcn_bias_relu_kernel<F_IN>
      <<<(int)(((long)N_NODES * F_IN + BT - 1) / BT), BT, 0, stream>>>(
          out, b2, (long)N_NODES * F_IN);
}